// ContrastiveGNN_661424963806
// MI455X (gfx1250) — compile-verified
//
#include <hip/hip_runtime.h>
#include <hip/hip_bf16.h>

typedef float v2f __attribute__((ext_vector_type(2)));
typedef float v8f __attribute__((ext_vector_type(8)));

#define N_NODES   100000
#define N_EDGES   1600000
#define N_FEAT    128
#define HIDDEN    128
#define EMB       256
#define N_GROUPS  16
#define N_GRAPHS  512
#define NPAD2     16     // float2 row padding: 2*(N+16) % 64 == 32 banks -> conflict-free b64

// Hardware fp32 atomic add (L2 atomic unit, no CAS loop).
__device__ __forceinline__ void gadd_f32(float* p, float v) {
    asm volatile("global_atomic_add_f32 %0, %1, off" :: "v"(p), "v"(v) : "memory");
}

// ---------------- degree / normalization ----------------
__global__ void k_init_deg(float* __restrict__ deg) {
    int i = blockIdx.x * blockDim.x + threadIdx.x;
    if (i < N_NODES) deg[i] = 1.0f;   // self loop
}

__global__ void k_count_deg(const int* __restrict__ dst, float* __restrict__ deg) {
    int e = blockIdx.x * blockDim.x + threadIdx.x;
    if (e < N_EDGES) gadd_f32(&deg[dst[e]], 1.0f);
}

__global__ void k_dinv(float* __restrict__ deg) {
    int i = blockIdx.x * blockDim.x + threadIdx.x;
    if (i < N_NODES) deg[i] = rsqrtf(deg[i]);
}

// ---------------- fp32 WMMA GEMM: H[M,N] = (relu?)X[M,128] @ W[128,N] ----------------
// Block = 8 waves = 128 M-rows. W staged in LDS as K-pair-interleaved float2
// (sW2[k/2][col] = {W[k][col], W[k+1][col]}) so a lane's 16x4 B fragment is ONE
// aligned ds_load_b64. Each wave:
//  - A-panel 16x128 f32 preloaded as 32 WMMA fragments (64 VGPRs)
//  - per 16-wide N-tile: burst b64 fragment loads, then accumulator-chained
//    V_WMMA_F32_16X16X4_F32 back-to-back (compiler double-buffers the waits).
__global__ __launch_bounds__(256) void k_gemm_wmma(
    const float* __restrict__ X, const float* __restrict__ W,
    float* __restrict__ H, int M, int N, int nShift /* log2(N) */,
    int applyRelu)
{
    extern __shared__ float2 sW2[];        // [64][N + NPAD2]
    const int K   = 128;
    const int NP2 = N + NPAD2;

    // ---- stage W into LDS, K-pair interleaved: one b64 store per (k/2, col) ----
    for (int idx = threadIdx.x; idx < (64 << nShift); idx += 256) {
        int k2  = idx >> nShift;
        int col = idx & (N - 1);
        float2 v;
        v.x = W[(size_t)(2 * k2)     * N + col];
        v.y = W[(size_t)(2 * k2 + 1) * N + col];
        sW2[(size_t)k2 * NP2 + col] = v;
    }
    __syncthreads();

    const int wave = threadIdx.x >> 5;
    const int lane = threadIdx.x & 31;
    const int row0 = blockIdx.x * 128 + wave * 16;

    const int r  = lane & 15;     // M index within tile (A) / N index (B,C)
    const int kh = lane >> 4;     // K half-select per ISA A/B fragment layout

    int arow = row0 + r;
    if (arow >= M) arow = M - 1;  // clamp; stores are guarded below

    // A fragments: lane<16 gets K = 4k+0,4k+1 ; lane>=16 gets K = 4k+2,4k+3
    const v2f* aptr = (const v2f*)(X + (size_t)arow * K + kh * 2);
    v2f afrag[32];
#pragma unroll
    for (int ks = 0; ks < 32; ++ks) {
        v2f t = aptr[ks * 2];
        if (applyRelu) { t.x = fmaxf(t.x, 0.0f); t.y = fmaxf(t.y, 0.0f); }
        afrag[ks] = t;
    }

    for (int nt = 0; nt < (N >> 4); ++nt) {
        const int col = (nt << 4) + r;
        // fragment ks lives at sW2[(2*ks + kh)][col]
        const float2* wp = sW2 + (size_t)kh * NP2 + col;

        v2f bfrag[32];
#pragma unroll
        for (int ks = 0; ks < 32; ++ks) {
            float2 t = wp[(size_t)(2 * ks) * NP2];   // single ds_load_b64
            bfrag[ks].x = t.x;
            bfrag[ks].y = t.y;
        }

        v8f acc = {};
#pragma unroll
        for (int ks = 0; ks < 32; ++ks) {
            acc = __builtin_amdgcn_wmma_f32_16x16x4_f32(
                false, afrag[ks], false, bfrag[ks], (short)0, acc, false, false);
        }

        // C/D layout: VGPR i, lane<16 -> row i, lane>=16 -> row i+8; col = lane&15
        const int rbase = row0 + kh * 8;
#pragma unroll
        for (int i = 0; i < 8; ++i) {
            const int rr = rbase + i;
            if (rr < M) H[(size_t)rr * N + col] = acc[i];
        }
    }
}

// ---------------- out[i] = H[i] * dinv[i]^2 + bias  (self-loop + bias init) ----------------
__global__ void k_self_bias(const float* __restrict__ H, const float* __restrict__ dinv,
                            const float* __restrict__ bias, float* __restrict__ out,
                            int cshift /* log2(N/4) */)
{
    size_t t = (size_t)blockIdx.x * blockDim.x + threadIdx.x;
    size_t node = t >> cshift;
    if (node >= N_NODES) return;
    int c = (int)(t & ((1u << cshift) - 1u));
    float di = dinv[node];
    float w  = di * di;
    float4 h = ((const float4*)H)[t];
    float4 b = ((const float4*)bias)[c];
    float4 o;
    o.x = h.x * w + b.x; o.y = h.y * w + b.y;
    o.z = h.z * w + b.z; o.w = h.w * w + b.w;
    ((float4*)out)[t] = o;
}

// ---------------- edge scatter: out[dst] += H[src] * dinv[src]*dinv[dst] ----------------
// One wave handles one edge's feature row (N=128) or half (N=256): fully coalesced
// float4 gathers, hardware fadd scatters.
__global__ void k_scatter(const int* __restrict__ src, const int* __restrict__ dst,
                          const float* __restrict__ dinv, const float* __restrict__ H,
                          float* __restrict__ out, int N, int cshift /* log2(N/4) */)
{
    size_t t = (size_t)blockIdx.x * blockDim.x + threadIdx.x;
    size_t e = t >> cshift;
    if (e >= N_EDGES) return;
    int c = (int)(t & ((1u << cshift) - 1u));
    int s = src[e], d = dst[e];
    float norm = dinv[s] * dinv[d];
    float4 h = ((const float4*)(H + (size_t)s * N))[c];
    float* o = out + (size_t)d * N + c * 4;
    gadd_f32(o + 0, h.x * norm);
    gadd_f32(o + 1, h.y * norm);
    gadd_f32(o + 2, h.z * norm);
    gadd_f32(o + 3, h.w * norm);
}

// ---------------- pooling ----------------
__global__ void k_zero(float* __restrict__ p, size_t n) {
    size_t t = (size_t)blockIdx.x * blockDim.x + threadIdx.x;
    if (t < n) p[t] = 0.0f;
}

__global__ void k_pool_sum(const int* __restrict__ batch, const float* __restrict__ h,
                           float* __restrict__ emb)
{
    size_t t = (size_t)blockIdx.x * blockDim.x + threadIdx.x;   // nodes * 64 chunks
    size_t node = t >> 6;
    if (node >= N_NODES) return;
    int c = (int)(t & 63);
    int g = batch[node];
    float4 v = ((const float4*)(h + node * EMB))[c];
    float* o = emb + (size_t)g * EMB + c * 4;
    gadd_f32(o + 0, v.x); gadd_f32(o + 1, v.y);
    gadd_f32(o + 2, v.z); gadd_f32(o + 3, v.w);
}

__global__ void k_pool_cnt(const int* __restrict__ batch, float* __restrict__ cnt) {
    int i = blockIdx.x * blockDim.x + threadIdx.x;
    if (i < N_NODES) gadd_f32(&cnt[batch[i]], 1.0f);
}

__global__ void k_pool_div(float* __restrict__ emb, const float* __restrict__ cnt) {
    int t = blockIdx.x * blockDim.x + threadIdx.x;
    if (t < N_GRAPHS * EMB) {
        int g = t >> 8;
        emb[t] = emb[t] / fmaxf(cnt[g], 1.0f);
    }
}

// ---------------- logits: [512,16] = emb[512,256] @ Wg[256,16] + bg ----------------
__global__ void k_logits(const float* __restrict__ emb, const float* __restrict__ Wg,
                         const float* __restrict__ bg, float* __restrict__ logits)
{
    int t = blockIdx.x * blockDim.x + threadIdx.x;
    if (t >= N_GRAPHS * N_GROUPS) return;
    int g = t >> 4, j = t & 15;
    const float* e = emb + (size_t)g * EMB;
    float s = bg[j];
#pragma unroll 8
    for (int k = 0; k < EMB; ++k) s = fmaf(e[k], Wg[k * N_GROUPS + j], s);
    logits[t] = s;
}

extern "C" void kernel_launch(void* const* d_in, const int* in_sizes, int n_in,
                              void* d_out, int out_size, void* d_ws, size_t ws_size,
                              hipStream_t stream)
{
    const float* x     = (const float*)d_in[0];
    const int*   ei    = (const int*)d_in[1];
    const int*   src   = ei;
    const int*   dst   = ei + N_EDGES;
    const int*   batch = (const int*)d_in[2];
    const float* W1    = (const float*)d_in[4];
    const float* b1    = (const float*)d_in[5];
    const float* W2    = (const float*)d_in[6];
    const float* b2    = (const float*)d_in[7];
    const float* Wg    = (const float*)d_in[8];
    const float* bg    = (const float*)d_in[9];

    float* outp   = (float*)d_out;
    float* emb    = outp;                                  // [512*256]
    float* logits = outp + (size_t)N_GRAPHS * EMB;         // [512*16]

    float* ws   = (float*)d_ws;
    float* dinv = ws;                                      // 100352
    float* cnt  = ws + 100352;                             // 512
    float* bufH = ws + 100864;                             // N_NODES*256 (GEMM out)
    float* bufO = bufH + (size_t)N_NODES * EMB;            // N_NODES*256 (aggregate)

    const int T = 256;
    // degree -> dinv
    k_init_deg <<<(N_NODES + T - 1) / T, T, 0, stream>>>(dinv);
    k_count_deg<<<(N_EDGES + T - 1) / T, T, 0, stream>>>(dst, dinv);
    k_dinv     <<<(N_NODES + T - 1) / T, T, 0, stream>>>(dinv);

    const int gemmBlocks = (N_NODES + 127) / 128;
    const size_t lds1 = (size_t)64 * (HIDDEN + NPAD2) * sizeof(float2);  //  73.7 KB
    const size_t lds2 = (size_t)64 * (EMB    + NPAD2) * sizeof(float2);  // 139.3 KB

    // conv1: H1 = x @ W1 ; out1 = agg(H1) + b1   (ReLU deferred to conv2 load)
    k_gemm_wmma<<<gemmBlocks, 256, lds1, stream>>>(x, W1, bufH, N_NODES, HIDDEN, 7, 0);
    {
        size_t tot = (size_t)N_NODES * (HIDDEN / 4);
        k_self_bias<<<(tot + T - 1) / T, T, 0, stream>>>(bufH, dinv, b1, bufO, 5);
        size_t st = (size_t)N_EDGES * (HIDDEN / 4);
        k_scatter<<<(st + T - 1) / T, T, 0, stream>>>(src, dst, dinv, bufH, bufO, HIDDEN, 5);
    }

    // conv2: H2 = relu(out1) @ W2 ; out2 = agg(H2) + b2
    k_gemm_wmma<<<gemmBlocks, 256, lds2, stream>>>(bufO, W2, bufH, N_NODES, EMB, 8, 1);
    {
        size_t tot = (size_t)N_NODES * (EMB / 4);
        k_self_bias<<<(tot + T - 1) / T, T, 0, stream>>>(bufH, dinv, b2, bufO, 6);
        size_t st = (size_t)N_EDGES * (EMB / 4);
        k_scatter<<<(st + T - 1) / T, T, 0, stream>>>(src, dst, dinv, bufH, bufO, EMB, 6);
    }

    // global mean pool -> emb
    k_zero<<<((size_t)N_GRAPHS * EMB + T - 1) / T, T, 0, stream>>>(emb, (size_t)N_GRAPHS * EMB);
    k_zero<<<(N_GRAPHS + T - 1) / T, T, 0, stream>>>(cnt, N_GRAPHS);
    {
        size_t tot = (size_t)N_NODES * (EMB / 4);
        k_pool_sum<<<(tot + T - 1) / T, T, 0, stream>>>(batch, bufO, emb);
    }
    k_pool_cnt<<<(N_NODES + T - 1) / T, T, 0, stream>>>(batch, cnt);
    k_pool_div<<<(N_GRAPHS * EMB + T - 1) / T, T, 0, stream>>>(emb, cnt);

    // group logits
    k_logits<<<(N_GRAPHS * N_GROUPS + T - 1) / T, T, 0, stream>>>(emb, Wg, bg, logits);
}